// TimeFeatureEmbedding_70798240907878
// MI455X (gfx1250) — compile-verified
//
#include <hip/hip_runtime.h>
#include <hip/hip_bf16.h>

typedef __attribute__((ext_vector_type(2))) float v2f;
typedef __attribute__((ext_vector_type(4))) float v4f;
typedef __attribute__((ext_vector_type(8))) float v8f;

#define D_MODEL 512
#define D_INP   4
#define S_REP   64
#define BT_TILE 16   // rows of (b,t) per block
#define NBT     3072 // 32*96

// y[bt, d] = sum_k x[bt,k] * W[d,k] + b[d]; out[bt, s, d] = y[bt, d]
__global__ __launch_bounds__(256) void tfe_wmma_bcast_kernel(
    const float* __restrict__ x,     // [NBT, 4]
    const float* __restrict__ W,     // [512, 4]  (row-major: W[d*4 + k])
    const float* __restrict__ bias,  // [512]
    float* __restrict__ out)         // [NBT, 64, 512]
{
    __shared__ float tile[BT_TILE * D_MODEL]; // 32 KB LDS staging of y-tile

    const int bt0  = blockIdx.x * BT_TILE;
    const int tid  = threadIdx.x;
    const int lane = tid & 31;
    const int wave = tid >> 5;       // 0..7, each wave covers 64 d-columns

    // ---------------- Phase 1: WMMA GEMM (K = 4 = D_INP, one WMMA per tile)
    // A layout (16x4 f32): lanes 0-15 hold M=lane, VGPR v -> K=v (+2 for lanes 16-31)
    const int half = lane >> 4;      // 0: K=0,1   1: K=2,3
    const int lm   = lane & 15;      // M for A / N for B
    v2f a = *(const v2f*)(x + (size_t)(bt0 + lm) * D_INP + 2 * half);

    #pragma unroll
    for (int j = 0; j < 4; ++j) {
        const int d0 = wave * 64 + j * 16;
        // B layout (4x16 f32): lane holds N=lm; VGPR v -> K = 2*half + v
        // B[k][n] = W[d0+n][k]  -> contiguous float2 per lane
        v2f bmat = *(const v2f*)(W + (size_t)(d0 + lm) * D_INP + 2 * half);

        v8f c = {};
        c = __builtin_amdgcn_wmma_f32_16x16x4_f32(
                /*neg_a=*/false, a, /*neg_b=*/false, bmat,
                /*c_mod=*/(short)0, c, /*reuse_a=*/false, /*reuse_b=*/false);

        // C/D layout: VGPR r, lane: M = r + 8*half, N = lm
        const float bv = bias[d0 + lm];
        #pragma unroll
        for (int r = 0; r < 8; ++r) {
            const int m = r + 8 * half;
            tile[m * D_MODEL + d0 + lm] = c[r] + bv;
        }
    }
    __syncthreads();

    // ---------------- Phase 2: broadcast 64 copies of each row (store-BW bound)
    // thread t: row = t>>4 (0..15); 8 float4s at columns (t&15 + 16k), k=0..7
    const int row = tid >> 4;
    const int c4  = tid & 15;
    v4f vals[8];
    #pragma unroll
    for (int k = 0; k < 8; ++k)
        vals[k] = *(const v4f*)(tile + row * D_MODEL + (c4 + 16 * k) * 4);

    float* obase = out + (size_t)(bt0 + row) * S_REP * D_MODEL;
    for (int s = 0; s < S_REP; ++s) {
        float* o = obase + (size_t)s * D_MODEL;
        #pragma unroll
        for (int k = 0; k < 8; ++k) {
            // output (384 MiB) exceeds L2 and is never re-read: non-temporal
            __builtin_nontemporal_store(vals[k], (v4f*)(o + (c4 + 16 * k) * 4));
        }
    }
}

extern "C" void kernel_launch(void* const* d_in, const int* in_sizes, int n_in,
                              void* d_out, int out_size, void* d_ws, size_t ws_size,
                              hipStream_t stream) {
    // setup_inputs order: x [32*96*4] f32, S (scalar, ==64), W [512*4] f32, b [512] f32
    const float* x    = (const float*)d_in[0];
    const float* W    = (const float*)d_in[2];
    const float* bias = (const float*)d_in[3];
    float* out        = (float*)d_out;

    const int blocks = NBT / BT_TILE; // 192
    tfe_wmma_bcast_kernel<<<blocks, 256, 0, stream>>>(x, W, bias, out);
}